// Cost_59476707115490
// MI455X (gfx1250) — compile-verified
//
#include <hip/hip_runtime.h>
#include <hip/hip_bf16.h>

// ---------------------------------------------------------------------------
// Cost-map kernel for MI455X (gfx1250).
// B=8, T=11 -> 80 frames of 384x384. Bandwidth-bound: ~472 MB of traffic,
// ~20us at 23.3 TB/s. One streaming pass fuses mask gen + mask store + count
// accumulation (exact integer counts -> deterministic). bev reads and mask
// writes are single-touch -> non-temporal cache policy (TH=NT) so the stream
// doesn't thrash the 192MB L2. Final decay-weighted reduction runs on
// V_WMMA_F32_16X16X4_F32 in a single wave32.
// ---------------------------------------------------------------------------

#define IMG 384
#define PIX (IMG * IMG)            // 147456
#define NBT 80                     // B * (T-1)
#define MASK_ELEMS ((size_t)NBT * PIX)  // 11,796,480
#define PIX_PER_THREAD 4
#define BLOCK 256

typedef float v2f __attribute__((ext_vector_type(2)));
typedef float v4f __attribute__((ext_vector_type(4)));
typedef float v8f __attribute__((ext_vector_type(8)));

// ---------------------------------------------------------------------------
// Kernel 0: per-(b,t) scalar params + zero the integer count buffer.
// params[bt*8] = {x, y, cos(yaw), sin(yaw), dy, dy_light}
// ---------------------------------------------------------------------------
__global__ void setup_params(const float* __restrict__ loc,
                             const float* __restrict__ yaw,
                             const float* __restrict__ speed,
                             float* __restrict__ params,
                             int* __restrict__ counts) {
  const int tid = threadIdx.x;
  for (int i = tid; i < NBT * 8; i += BLOCK) counts[i] = 0;
  if (tid < NBT) {
    const int b = tid / 10;
    const int tt = (tid % 10) + 1;          // frame index 1..10
    const float yaw0 = yaw[b * 11];
    const float c0 = cosf(yaw0), s0 = sinf(yaw0);
    const float relx = loc[(b * 11 + tt) * 2 + 0] - loc[(b * 11) * 2 + 0];
    const float rely = loc[(b * 11 + tt) * 2 + 1] - loc[(b * 11) * 2 + 1];
    const float x = c0 * relx + s0 * rely;
    const float y = -s0 * relx + c0 * rely;
    const float yw = yaw[b * 11 + tt] - yaw0;
    const float sp = speed[b * 11 + tt];
    const float dy = 1.5f * (fmaxf(10.0f, sp) + 4.9f) + 1.0f;
    const float dyl = sp * 0.5f + 4.9f * 3.0f;
    float* P = params + tid * 8;
    P[0] = x; P[1] = y; P[2] = cosf(yw); P[3] = sinf(yw); P[4] = dy; P[5] = dyl;
  }
}

// ---------------------------------------------------------------------------
// Kernel 1: streaming pass. grid = (PIX/(BLOCK*4), 80).
// Each thread: 4 consecutive pixels (same row since 384 % 4 == 0).
// Non-temporal b128 bev loads + non-temporal mask stores (single-touch data).
// Accumulates 7 integer counts (exact, deterministic).
// ---------------------------------------------------------------------------
__global__ void mask_cost_kernel(const float* __restrict__ bev,
                                 const float* __restrict__ params,
                                 int* __restrict__ counts,
                                 float* __restrict__ out) {
  const int bt = blockIdx.y;
  const int b = bt / 10;
  const int t = bt % 10;
  const int tid = threadIdx.x;
  const int p0 = (blockIdx.x * BLOCK + tid) * PIX_PER_THREAD;
  const int row = p0 / IMG;
  const int j0 = p0 % IMG;

  const float* P = params + bt * 8;
  const float x = P[0], y = P[1], c = P[2], s = P[3], dy = P[4], dyl = P[5];
  const float Yc = ((float)row - 192.0f) * 0.2f;   // ys[row] = (row-192)/5

  // Load 7 bev channels (1..7) for this frame: 16B-aligned NT b128 streams.
  const size_t bevFrame = (size_t)(b * 11 + t + 1) * 8 * PIX;
  float chv[7][4];
#pragma unroll
  for (int cc = 0; cc < 7; ++cc) {
    const v4f v = __builtin_nontemporal_load(
        (const v4f*)(bev + bevFrame + (size_t)(cc + 1) * PIX + p0));
    chv[cc][0] = v[0]; chv[cc][1] = v[1]; chv[cc][2] = v[2]; chv[cc][3] = v[3];
  }

  int acc[7] = {0, 0, 0, 0, 0, 0, 0};
  float* const out_car   = out + 7;
  float* const out_side  = out + 7 + MASK_ELEMS;
  float* const out_light = out + 7 + 2 * MASK_ELEMS;
  const size_t pixBase = (size_t)bt * PIX + p0;

#pragma unroll
  for (int u = 0; u < PIX_PER_THREAD; ++u) {
    const float X = ((float)(j0 + u) - 192.0f) * 0.2f;
    const float rx = X - x;
    const float ry = Yc - y;
    const float ax = c * rx + s * ry;
    const float ay = -s * rx + c * ry;
    const float rxl = rx - 12.0f;       // X - (x + LIGHT_OFF_X)
    const float ryl = ry - 3.25f;       // Y - (y + LIGHT_OFF_Y)
    const float axl = c * rxl + s * ryl;
    const float ayl = -s * rxl + c * ryl;

    const int mc = (fabsf(ax) <= 5.05f && ay >= 0.0f && ay <= dy) ? 1 : 0;
    const int ms = (fabsf(ax) <= 1.155f && fabsf(ay) <= 2.695f) ? 1 : 0;
    const int ml = (fabsf(axl) <= 3.1f && ayl >= 0.0f && ayl <= dyl) ? 1 : 0;

    __builtin_nontemporal_store((float)mc, out_car + pixBase + u);
    __builtin_nontemporal_store((float)ms, out_side + pixBase + u);
    __builtin_nontemporal_store((float)ml, out_light + pixBase + u);

    const int b1 = chv[0][u] > 0.5f;  // lane
    const int b2 = chv[1][u] > 0.5f;  // vehicle
    const int b3 = chv[2][u] > 0.5f;  // green
    const int b4 = chv[3][u] > 0.5f;  // yellow
    const int b5 = chv[4][u] > 0.5f;  // red
    const int b6 = chv[5][u] > 0.5f;  // pedestrian
    const int b7 = chv[6][u] > 0.5f;  // offroad
    acc[0] += b1 & ms;
    acc[1] += b2 & mc;
    acc[2] += b3 & ml;
    acc[3] += b4 & ml;
    acc[4] += b5 & ml;
    acc[5] += b6 & mc;
    acc[6] += b7 & ms;
  }

  // wave32 reduction, then LDS atomics, then global integer atomics.
#pragma unroll
  for (int off = 16; off > 0; off >>= 1) {
#pragma unroll
    for (int k = 0; k < 7; ++k) acc[k] += __shfl_down(acc[k], off, 32);
  }

  __shared__ int sacc[7];
  if (tid < 7) sacc[tid] = 0;
  __syncthreads();
  if ((tid & 31) == 0) {
#pragma unroll
    for (int k = 0; k < 7; ++k)
      if (acc[k]) atomicAdd(&sacc[k], acc[k]);
  }
  __syncthreads();
  if (tid < 7) atomicAdd(&counts[bt * 8 + tid], sacc[tid]);
}

// ---------------------------------------------------------------------------
// Kernel 2: cost_m = sum_bt decay[t(bt)] * counts[bt][m], via
// V_WMMA_F32_16X16X4_F32 in one wave. A(16x4) = counts (row = cost index),
// B(4x16) = decay broadcast across columns; C accumulates over 20 steps.
// D[m,n] is independent of n; lane 0 holds D[0..7, 0] in its 8 C VGPRs.
// Decay comes from an f32 literal table (no divergent pow loop).
// ---------------------------------------------------------------------------
__device__ __forceinline__ float decay_of(int t) {
  const float DTAB[10] = {
      1.0f,            0.97f,           0.9409f,         0.912673f,
      0.88529281f,     0.8587340257f,   0.83297190f,     0.80798274f,
      0.78374326f,     0.76023096f};
  return DTAB[t];
}

__global__ void final_reduce(const int* __restrict__ counts,
                             float* __restrict__ out) {
  const int lane = threadIdx.x;        // 0..31, EXEC all ones at the WMMA
  const int m = lane & 15;             // matrix row held by this lane
  const int k0 = (lane < 16) ? 0 : 2;  // K pair per lane half (A/B f32 layout)

  v8f acc = {};
  for (int kk = 0; kk < NBT; kk += 4) {
    const int i0 = kk + k0;
    const int i1 = kk + k0 + 1;
    const float a0 = (m < 7) ? (float)counts[i0 * 8 + m] : 0.0f;
    const float a1 = (m < 7) ? (float)counts[i1 * 8 + m] : 0.0f;
    v2f A = {a0, a1};
    v2f B = {decay_of(i0 % 10), decay_of(i1 % 10)};
    acc = __builtin_amdgcn_wmma_f32_16x16x4_f32(
        /*neg_a=*/false, A, /*neg_b=*/false, B,
        /*c_mod=*/(short)0, acc, /*reuse_a=*/false, /*reuse_b=*/false);
  }

  if (lane == 0) {
#pragma unroll
    for (int r = 0; r < 7; ++r) out[r] = acc[r];  // D[r, 0]
  }
}

// ---------------------------------------------------------------------------
extern "C" void kernel_launch(void* const* d_in, const int* in_sizes, int n_in,
                              void* d_out, int out_size, void* d_ws, size_t ws_size,
                              hipStream_t stream) {
  const float* loc   = (const float*)d_in[0];   // (8,11,2)
  const float* yaw   = (const float*)d_in[1];   // (8,11,1)
  const float* speed = (const float*)d_in[2];   // (8,11,1)
  const float* bev   = (const float*)d_in[3];   // (8,11,8,384,384)
  float* out = (float*)d_out;                   // 7 scalars + 3 masks

  float* params = (float*)d_ws;                 // 80*8 floats
  int*   counts = (int*)d_ws + NBT * 8;         // 80*8 ints

  setup_params<<<1, BLOCK, 0, stream>>>(loc, yaw, speed, params, counts);

  dim3 grid(PIX / (BLOCK * PIX_PER_THREAD), NBT);  // (144, 80)
  mask_cost_kernel<<<grid, BLOCK, 0, stream>>>(bev, params, counts, out);

  final_reduce<<<1, 32, 0, stream>>>(counts, out);
}